// GraphSAGELayer_2954937499913
// MI455X (gfx1250) — compile-verified
//
#include <hip/hip_runtime.h>

typedef __attribute__((ext_vector_type(2))) float v2f;
typedef __attribute__((ext_vector_type(8))) float v8f;

#define DF   64     // feature dim
#define K2   128    // 2*DF (concat input to linear)
#define WPAD 132    // 128 + 4 pad -> conflict-free LDS column access
#define OPAD 68     // 64 + 4 pad

// ---------------------------------------------------------------------------
// Kernel 0: zero the workspace (agg[N*64] ++ deg[N]) -- harness poisons d_ws.
// ---------------------------------------------------------------------------
__global__ void zero_ws_kernel(float* __restrict__ ws, long long n) {
  long long i = (long long)blockIdx.x * blockDim.x + threadIdx.x;
  if (i < n) ws[i] = 0.0f;
}

// ---------------------------------------------------------------------------
// Kernel 1: edge scatter (mean-agg numerator + degree), native fp32 atomics.
// 64 threads per edge: coalesced 256B read of x[src], coalesced atomic adds.
// x and agg are L2-resident (25.6 MB each << 192 MB L2).
// ---------------------------------------------------------------------------
__global__ void __launch_bounds__(256)
scatter_sum_kernel(const float* __restrict__ x,
                   const long long* __restrict__ ei,
                   float* __restrict__ agg,
                   float* __restrict__ deg,
                   int N, long long E) {
  long long t = (long long)blockIdx.x * blockDim.x + threadIdx.x;
  long long e = t >> 6;          // edge id
  int d = (int)(t & 63);         // feature id
  if (e >= E) return;
  int dst = (int)ei[e];          // edge_index[0][e]
  int src = (int)ei[E + e];      // edge_index[1][e]
  float val = x[(long long)src * DF + d];
  __hip_atomic_fetch_add(&agg[(long long)dst * DF + d], val,
                         __ATOMIC_RELAXED, __HIP_MEMORY_SCOPE_AGENT);
  if (d == 0) {
    __hip_atomic_fetch_add(&deg[dst], 1.0f,
                           __ATOMIC_RELAXED, __HIP_MEMORY_SCOPE_AGENT);
  }
}

// ---------------------------------------------------------------------------
// Kernel 2: fused  out = LN(relu([x | agg/deg] @ W^T + b)) * gamma + beta
// Block = 128 threads (4 x wave32) handles 64 nodes.
// Wave w computes rows [16w,16w+16) x all 64 output cols via
// v_wmma_f32_16x16x4_f32, K-loop of 32 steps over the 128-wide concat input.
// ---------------------------------------------------------------------------
__global__ void __launch_bounds__(128)
sage_gemm_ln_kernel(const float* __restrict__ x,
                    const float* __restrict__ agg,
                    const float* __restrict__ deg,
                    const float* __restrict__ W,     // [64][128]
                    const float* __restrict__ bias,  // [64]
                    const float* __restrict__ gamma, // [64]
                    const float* __restrict__ beta,  // [64]
                    float* __restrict__ out,         // [N][64]
                    int N) {
  __shared__ float w_lds[64 * WPAD];    // 33.8 KB, padded
  __shared__ float out_lds[64 * OPAD];  // 17.4 KB, padded
  __shared__ float invd_lds[64];
  __shared__ float mean_lds[64];
  __shared__ float rstd_lds[64];

  const int tid = threadIdx.x;
  const int nodeBase = blockIdx.x * 64;

  // Stage W into padded LDS: coalesced global reads, conflict-free stores.
  for (int idx = tid; idx < 64 * K2; idx += 128) {
    int n = idx >> 7, k = idx & 127;
    w_lds[n * WPAD + k] = W[idx];
  }
  // Per-row 1/max(deg,1)
  if (tid < 64) {
    int node = nodeBase + tid;
    int cn = node < N ? node : N - 1;
    invd_lds[tid] = 1.0f / fmaxf(deg[cn], 1.0f);
  }
  __syncthreads();

  const int wave = tid >> 5;
  const int lane = tid & 31;
  const int lr   = lane & 15;   // row-in-tile (A), col-in-tile (B/C/D)
  const int hi   = lane >> 4;   // half-wave select: K offset / M offset
  const int koff = 2 * hi;      // A/B: lanes 16-31 hold K+2,K+3

  int row = nodeBase + wave * 16 + lr;
  if (row >= N) row = N - 1;                     // clamp; stores are guarded
  const float invd = invd_lds[wave * 16 + lr];

  v8f acc0 = {}, acc1 = {}, acc2 = {}, acc3 = {};

  // ---- K-phase 1: k in [0,64) -> A comes from x (no divergence: boundary
  //      64 is a multiple of the k-step 4, so each step is wholly one side).
  for (int kk = 0; kk < 64; kk += 4) {
    const int k0 = kk + koff;
    const float2 ax = *(const float2*)(x + (long long)row * DF + k0);
    v2f a; a.x = ax.x; a.y = ax.y;

    const float2 w0 = *(const float2*)&w_lds[( 0 + lr) * WPAD + k0];
    const float2 w1 = *(const float2*)&w_lds[(16 + lr) * WPAD + k0];
    const float2 w2 = *(const float2*)&w_lds[(32 + lr) * WPAD + k0];
    const float2 w3 = *(const float2*)&w_lds[(48 + lr) * WPAD + k0];
    v2f b0; b0.x = w0.x; b0.y = w0.y;
    v2f b1; b1.x = w1.x; b1.y = w1.y;
    v2f b2; b2.x = w2.x; b2.y = w2.y;
    v2f b3; b3.x = w3.x; b3.y = w3.y;

    acc0 = __builtin_amdgcn_wmma_f32_16x16x4_f32(false, a, false, b0, (short)0, acc0, false, false);
    acc1 = __builtin_amdgcn_wmma_f32_16x16x4_f32(false, a, false, b1, (short)0, acc1, false, false);
    acc2 = __builtin_amdgcn_wmma_f32_16x16x4_f32(false, a, false, b2, (short)0, acc2, false, false);
    acc3 = __builtin_amdgcn_wmma_f32_16x16x4_f32(false, a, false, b3, (short)0, acc3, false, false);
  }

  // ---- K-phase 2: k in [64,128) -> A comes from agg * (1/deg).
  for (int kk = 64; kk < K2; kk += 4) {
    const int k0 = kk + koff;
    const float2 ag = *(const float2*)(agg + (long long)row * DF + (k0 - 64));
    v2f a; a.x = ag.x * invd; a.y = ag.y * invd;

    const float2 w0 = *(const float2*)&w_lds[( 0 + lr) * WPAD + k0];
    const float2 w1 = *(const float2*)&w_lds[(16 + lr) * WPAD + k0];
    const float2 w2 = *(const float2*)&w_lds[(32 + lr) * WPAD + k0];
    const float2 w3 = *(const float2*)&w_lds[(48 + lr) * WPAD + k0];
    v2f b0; b0.x = w0.x; b0.y = w0.y;
    v2f b1; b1.x = w1.x; b1.y = w1.y;
    v2f b2; b2.x = w2.x; b2.y = w2.y;
    v2f b3; b3.x = w3.x; b3.y = w3.y;

    acc0 = __builtin_amdgcn_wmma_f32_16x16x4_f32(false, a, false, b0, (short)0, acc0, false, false);
    acc1 = __builtin_amdgcn_wmma_f32_16x16x4_f32(false, a, false, b1, (short)0, acc1, false, false);
    acc2 = __builtin_amdgcn_wmma_f32_16x16x4_f32(false, a, false, b2, (short)0, acc2, false, false);
    acc3 = __builtin_amdgcn_wmma_f32_16x16x4_f32(false, a, false, b3, (short)0, acc3, false, false);
  }

  // Bias + ReLU, spill tile to LDS for the row-wise LayerNorm reduction.
  // C/D layout: VGPR r holds M = r + 8*hi, N = lr.
  const float bs0 = bias[ 0 + lr];
  const float bs1 = bias[16 + lr];
  const float bs2 = bias[32 + lr];
  const float bs3 = bias[48 + lr];
#pragma unroll
  for (int r = 0; r < 8; ++r) {
    const int lrow = wave * 16 + 8 * hi + r;
    out_lds[lrow * OPAD +  0 + lr] = fmaxf(acc0[r] + bs0, 0.0f);
    out_lds[lrow * OPAD + 16 + lr] = fmaxf(acc1[r] + bs1, 0.0f);
    out_lds[lrow * OPAD + 32 + lr] = fmaxf(acc2[r] + bs2, 0.0f);
    out_lds[lrow * OPAD + 48 + lr] = fmaxf(acc3[r] + bs3, 0.0f);
  }
  __syncthreads();

  // LayerNorm statistics: one thread per row.
  if (tid < 64) {
    float s = 0.0f, ss = 0.0f;
#pragma unroll 8
    for (int c = 0; c < 64; ++c) {
      float v = out_lds[tid * OPAD + c];
      s += v; ss += v * v;
    }
    const float mu  = s * (1.0f / 64.0f);
    const float var = ss * (1.0f / 64.0f) - mu * mu;
    mean_lds[tid] = mu;
    rstd_lds[tid] = rsqrtf(var + 1e-5f);
  }
  __syncthreads();

  // Normalize + affine, coalesced stores. col == tid&63 for every iteration.
  const int c = tid & 63;
  const float gm = gamma[c];
  const float bt = beta[c];
  for (int idx = tid; idx < 64 * 64; idx += 128) {
    const int r = idx >> 6;
    const int node = nodeBase + r;
    if (node < N) {
      const float v = out_lds[r * OPAD + c];
      out[(long long)node * DF + c] = (v - mean_lds[r]) * rstd_lds[r] * gm + bt;
    }
  }
}

// ---------------------------------------------------------------------------
extern "C" void kernel_launch(void* const* d_in, const int* in_sizes, int n_in,
                              void* d_out, int out_size, void* d_ws, size_t ws_size,
                              hipStream_t stream) {
  const float*     x     = (const float*)d_in[0];
  const long long* ei    = (const long long*)d_in[1];   // int64 [2][E]
  const float*     W     = (const float*)d_in[2];
  const float*     bias  = (const float*)d_in[3];
  const float*     gamma = (const float*)d_in[4];
  const float*     beta  = (const float*)d_in[5];
  float*           out   = (float*)d_out;

  const int       N = in_sizes[0] / DF;
  const long long E = (long long)in_sizes[1] / 2;

  float* agg = (float*)d_ws;                 // [N][64]
  float* deg = agg + (long long)N * DF;      // [N]

  const long long zn = (long long)N * (DF + 1);
  zero_ws_kernel<<<(unsigned)((zn + 255) / 256), 256, 0, stream>>>(agg, zn);

  const long long st = E * 64;
  scatter_sum_kernel<<<(unsigned)((st + 255) / 256), 256, 0, stream>>>(
      x, ei, agg, deg, N, E);

  const int nb = (N + 63) / 64;
  sage_gemm_ln_kernel<<<nb, 128, 0, stream>>>(
      x, agg, deg, W, bias, gamma, beta, out, N);
}